// EnsembleUncertaintyDistance_53824530153580
// MI455X (gfx1250) — compile-verified
//
#include <hip/hip_runtime.h>
#include <hip/hip_bf16.h>

typedef __attribute__((ext_vector_type(2))) float v2f;
typedef __attribute__((ext_vector_type(8))) float v8f;

#define EMB 512
#define NM  5
#define HID 64
#define NQ  2048
#define NP  256

// -----------------------------------------------------------------------------
// Stage 1: out[m][row][h] = sum_d X[row][d] * W1[m][h][dOff + d]  (+ b1[m][h])
// One wave (32 lanes) computes a 16(row) x 64(h) strip for one model using
// V_WMMA_F32_16X16X4_F32, K-loop over EMB in steps of 4.
// -----------------------------------------------------------------------------
__global__ __launch_bounds__(32)
void hidden_gemm_wmma(const float* __restrict__ X,     // [rows][EMB]
                      const float* __restrict__ W1,    // [NM][HID][2*EMB]
                      const float* __restrict__ b1,    // [NM][HID] or nullptr
                      float* __restrict__ out,         // [NM][rows][HID]
                      int dOff, int rows)
{
    const int q0   = blockIdx.x * 16;
    const int m    = blockIdx.y;
    const int lane = threadIdx.x;
    const int lo   = lane & 15;     // row (A) / col (B,D)
    const int hi   = lane >> 4;     // selects K pair {0,1} vs {2,3}

    const float* Xrow  = X + (size_t)(q0 + lo) * EMB + 2 * hi;
    const float* Wbase = W1 + (size_t)m * HID * (2 * EMB) + dOff + 2 * hi;

    v8f acc[4];
    acc[0] = (v8f){}; acc[1] = (v8f){}; acc[2] = (v8f){}; acc[3] = (v8f){};

    #pragma unroll 4
    for (int k4 = 0; k4 < EMB; k4 += 4) {
        // A fragment: lane l<16 -> X[q0+l][k4+0..1], lane l>=16 -> X[q0+l-16][k4+2..3]
        v2f a = *(const v2f*)(Xrow + k4);
        #pragma unroll
        for (int t = 0; t < 4; ++t) {
            // B fragment: B[k][n] = W1[m][h0+n][k]; same {0,1}/{2,3} split as A.
            v2f b = *(const v2f*)(Wbase + (size_t)(t * 16 + lo) * (2 * EMB) + k4);
            acc[t] = __builtin_amdgcn_wmma_f32_16x16x4_f32(
                /*neg_a=*/false, a, /*neg_b=*/false, b,
                /*c_mod=*/(short)0, acc[t], /*reuse_a=*/false, /*reuse_b=*/false);
        }
    }

    // D layout: VGPR r holds row (r + 8*hi), column lo.
    #pragma unroll
    for (int t = 0; t < 4; ++t) {
        const int col  = t * 16 + lo;
        const float bv = b1 ? b1[m * HID + col] : 0.0f;
        #pragma unroll
        for (int r = 0; r < 8; ++r) {
            const int row = q0 + r + 8 * hi;
            out[((size_t)m * rows + row) * HID + col] = acc[t][r] + bv;
        }
    }
}

// -----------------------------------------------------------------------------
// Stage 2: fused relu-reduce over h, plus mean/std(ddof=1)/exp across models.
// Block = 256 threads = 16q x 16p tile. LDS tiles padded (+1) to avoid
// stride-64 bank conflicts on the 64-bank LDS.
// -----------------------------------------------------------------------------
__global__ __launch_bounds__(256)
void combine_kernel(const float* __restrict__ hq,  // [NM][NQ][HID]
                    const float* __restrict__ hp,  // [NM][NP][HID], b1 folded in
                    const float* __restrict__ W2,  // [NM][HID]
                    const float* __restrict__ b2,  // [NM]
                    float* __restrict__ outp)      // [NQ][NP]
{
    __shared__ float s_hq[NM][16][HID + 1];
    __shared__ float s_hp[NM][16][HID + 1];
    __shared__ float s_w2[NM][HID];

    const int p0  = blockIdx.x * 16;
    const int q0  = blockIdx.y * 16;
    const int tid = threadIdx.x;

    for (int idx = tid; idx < NM * 16 * HID; idx += 256) {
        const int m   = idx / (16 * HID);
        const int rem = idx % (16 * HID);
        const int r   = rem / HID;
        const int h   = rem % HID;
        s_hq[m][r][h] = hq[((size_t)m * NQ + q0 + r) * HID + h];
        s_hp[m][r][h] = hp[((size_t)m * NP + p0 + r) * HID + h];
    }
    for (int idx = tid; idx < NM * HID; idx += 256)
        s_w2[idx / HID][idx % HID] = W2[idx];
    __syncthreads();

    const int q = tid & 15;
    const int p = tid >> 4;

    float s1 = 0.0f, s2 = 0.0f;
    #pragma unroll
    for (int m = 0; m < NM; ++m) {
        float om = b2[m];
        const float* __restrict__ aq = s_hq[m][q];
        const float* __restrict__ bp = s_hp[m][p];
        const float* __restrict__ w  = s_w2[m];
        #pragma unroll
        for (int h = 0; h < HID; ++h)
            om = fmaf(fmaxf(aq[h] + bp[h], 0.0f), w[h], om);
        s1 += om;
        s2 = fmaf(om, om, s2);
    }

    const float mean = s1 * (1.0f / NM);
    float var = (s2 - (float)NM * mean * mean) * (1.0f / (NM - 1));
    var = fmaxf(var, 0.0f);
    const float sd = sqrtf(var);
    outp[(size_t)(q0 + q) * NP + (p0 + p)] = mean * __expf(-sd);
}

// -----------------------------------------------------------------------------
extern "C" void kernel_launch(void* const* d_in, const int* in_sizes, int n_in,
                              void* d_out, int out_size, void* d_ws, size_t ws_size,
                              hipStream_t stream) {
    const float* Q  = (const float*)d_in[0];   // (2048, 512)
    const float* P  = (const float*)d_in[1];   // (256, 512)
    const float* W1 = (const float*)d_in[2];   // (5, 64, 1024)
    const float* b1 = (const float*)d_in[3];   // (5, 64)
    const float* W2 = (const float*)d_in[4];   // (5, 64)
    const float* b2 = (const float*)d_in[5];   // (5,)
    float* out = (float*)d_out;                // (2048, 256)

    float* hq = (float*)d_ws;                          // NM*NQ*HID floats
    float* hp = hq + (size_t)NM * NQ * HID;            // NM*NP*HID floats

    hidden_gemm_wmma<<<dim3(NQ / 16, NM), 32, 0, stream>>>(Q, W1, nullptr, hq, 0,   NQ);
    hidden_gemm_wmma<<<dim3(NP / 16, NM), 32, 0, stream>>>(P, W1, b1,      hp, EMB, NP);
    combine_kernel<<<dim3(NP / 16, NQ / 16), 256, 0, stream>>>(hq, hp, W2, b2, out);
}